// DeltaMemoryModule_3204045603524
// MI455X (gfx1250) — compile-verified
//
#include <hip/hip_runtime.h>

typedef __attribute__((ext_vector_type(2))) float v2f;
typedef __attribute__((ext_vector_type(4))) float v4f;
typedef __attribute__((ext_vector_type(8))) float v8f;

#define EPS 1e-6f
#define PROW 20   // padded LDS row stride (floats): 80B, 16B-aligned, spreads banks

// One wave32 per batch. M (64x64 f32) lives in 4x4 grid of WMMA f32 accumulators.
__global__ __launch_bounds__(32) void delta_mem_kernel(
    const float* __restrict__ h, const float* __restrict__ W,
    const float* __restrict__ bias, float* __restrict__ out, int L)
{
    const int b    = blockIdx.x;
    const int lane = threadIdx.x;       // 0..31
    const int c    = lane & 15;         // column within a 16-block
    const bool lo  = lane < 16;

    __shared__ __align__(16) float lds_k[64];
    __shared__ __align__(16) float lds_u[64];
    __shared__ __align__(16) float lds_p[64 * PROW];

    v8f acc[4][4];
    {
        v8f z = {0.f,0.f,0.f,0.f,0.f,0.f,0.f,0.f};
        #pragma unroll
        for (int i = 0; i < 4; ++i)
            #pragma unroll
            for (int j = 0; j < 4; ++j)
                acc[i][j] = z;
    }

    const float* hb = h + (size_t)b * (size_t)L * 64;
    const int rbase = lo ? 0 : 8;       // row offset of this half-wave in C layout

    for (int t = 0; t < L; ++t) {
        const bool update = (t < L - 1);

        if (t + 8 < L)  // stream-ahead into caches (global_prefetch_b8)
            __builtin_prefetch(hb + (size_t)(t + 8) * 64 + lane * 2, 0, 1);

        // ---- load k_t (2 floats/lane, coalesced), compute ||k||^2 ----
        v2f kv = *(const v2f*)(hb + (size_t)t * 64 + lane * 2);
        float k2 = kv.x * kv.x + kv.y * kv.y;
        #pragma unroll
        for (int m = 16; m >= 1; m >>= 1) k2 += __shfl_xor(k2, m, 32);
        const float denom = k2 + EPS;

        *(v2f*)(&lds_k[lane * 2]) = kv;
        __syncthreads();

        float kc[4];                     // k[16*bj + c], needed by every lane
        #pragma unroll
        for (int bj = 0; bj < 4; ++bj) kc[bj] = lds_k[bj * 16 + c];
        const float kr0 = lds_k[lane];
        const float kr1 = lds_k[lane + 32];

        // ---- v_pred = M @ k : per-lane column partials straight out of the
        //      WMMA C-layout, transposed through LDS ----
        #pragma unroll
        for (int bi = 0; bi < 4; ++bi) {
            #pragma unroll
            for (int v = 0; v < 8; ++v) {
                float p =      acc[bi][0][v] * kc[0];
                p = fmaf(acc[bi][1][v], kc[1], p);
                p = fmaf(acc[bi][2][v], kc[2], p);
                p = fmaf(acc[bi][3][v], kc[3], p);
                lds_p[(bi * 16 + v + rbase) * PROW + c] = p;
            }
        }
        __syncthreads();

        float vp0 = 0.f, vp1 = 0.f;      // rows: lane, lane+32
        {
            const float* r0 = &lds_p[lane * PROW];
            const float* r1 = &lds_p[(lane + 32) * PROW];
            #pragma unroll
            for (int q = 0; q < 16; q += 4) {
                v4f a = *(const v4f*)(r0 + q);
                v4f d = *(const v4f*)(r1 + q);
                vp0 += (a.x + a.y) + (a.z + a.w);
                vp1 += (d.x + d.y) + (d.z + d.w);
            }
        }

        if (!update) {                   // t == L-1: read = M @ h_last
            lds_u[lane]      = vp0;
            lds_u[lane + 32] = vp1;
            __syncthreads();
            break;
        }

        // ---- delta-rule direction + Hopfield gate ----
        float u0 = kr0 - vp0 / denom;
        float u1 = kr1 - vp1 / denom;
        float s1 = u0 * vp0 + u1 * vp1;  // <M, dM> = u . v_pred
        float s2 = u0 * u0 + u1 * u1;    // ||u||^2
        #pragma unroll
        for (int m = 16; m >= 1; m >>= 1) {
            s1 += __shfl_xor(s1, m, 32);
            s2 += __shfl_xor(s2, m, 32);
        }
        // accept iff ||M+dM||^2 > ||M||^2  <=>  2<M,dM> + ||dM||^2 > 0
        const float gate = (2.0f * s1 + s2 * k2 > 0.0f) ? 1.0f : 0.0f;
        u0 *= gate; u1 *= gate;          // fold gate: EXEC stays all-1s for WMMA

        lds_u[lane]      = u0;
        lds_u[lane + 32] = u1;
        __syncthreads();

        // ---- rank-1 update M += u_g * k^T via 16x V_WMMA_F32_16X16X4_F32 ----
        // A (16x4): col K=0 = u chunk; B (4x16): row K=0 = k chunk; rest zero.
        v2f A[4], Bv[4];
        #pragma unroll
        for (int i = 0; i < 4; ++i) {
            float uv = lds_u[i * 16 + c];
            A[i].x  = lo ? uv    : 0.0f;  A[i].y  = 0.0f;
            Bv[i].x = lo ? kc[i] : 0.0f;  Bv[i].y = 0.0f;
        }
        __syncthreads();

        #pragma unroll
        for (int bi = 0; bi < 4; ++bi)
            #pragma unroll
            for (int bj = 0; bj < 4; ++bj)
                acc[bi][bj] = __builtin_amdgcn_wmma_f32_16x16x4_f32(
                    false, A[bi], false, Bv[bj],
                    (short)0, acc[bi][bj], false, false);
    }

    // ---- out[b,i] = sum_j W[i,j]*read[j] + bias[i]; rows i = lane, lane+32 ----
    float o0 = bias[lane];
    float o1 = bias[lane + 32];
    const float* w0 = W + lane * 64;
    const float* w1 = W + (lane + 32) * 64;
    #pragma unroll 8
    for (int j = 0; j < 64; ++j) {
        float rj = lds_u[j];
        o0 = fmaf(w0[j], rj, o0);
        o1 = fmaf(w1[j], rj, o1);
    }
    out[b * 64 + lane]      = o0;
    out[b * 64 + lane + 32] = o1;
}

extern "C" void kernel_launch(void* const* d_in, const int* in_sizes, int n_in,
                              void* d_out, int out_size, void* d_ws, size_t ws_size,
                              hipStream_t stream) {
    (void)n_in; (void)d_ws; (void)ws_size;
    const float* h    = (const float*)d_in[0];  // [B, L, 64] f32
    const float* W    = (const float*)d_in[1];  // [64, 64]   f32
    const float* bias = (const float*)d_in[2];  // [64]       f32
    float* out = (float*)d_out;                 // [B, 64]    f32

    const int H = 64;
    const int B = out_size / H;                 // 256
    const int L = in_sizes[0] / (B * H);        // 2048

    delta_mem_kernel<<<dim3(B), dim3(32), 0, stream>>>(h, W, bias, out, L);
}